// DWT_1D_66924180407263
// MI455X (gfx1250) — compile-verified
//
#include <hip/hip_runtime.h>
#include <cmath>

// DWT_1D on MI455X (gfx1250): banded strided-conv formulation of the
// einsum('ncl,kl->nck') reference, computed with V_WMMA_F32_16X16X4_F32.
//
// Memory-bound: 32 MiB in + 32 MiB out -> ~2.9 us roofline at 23.3 TB/s.
// Round-2 change: hoist the edge predication out of the load loop (it is
// wave-uniform, only the first/last column tile can go out of range), so the
// interior path issues all 10 global_load_b64 back-to-back (one address +
// immediate offsets) before the WMMA chain, instead of the round-1
// load -> s_wait_loadcnt 0 -> wmma serialization.

typedef float v2f __attribute__((ext_vector_type(2)));
typedef float v8f __attribute__((ext_vector_type(8)));

__global__ __launch_bounds__(256) void dwt1d_wmma(const float* __restrict__ x,
                                                  const float* __restrict__ mlo,
                                                  const float* __restrict__ mhi,
                                                  float* __restrict__ out,
                                                  int R, int L, int Kout) {
  // Extract the 8 filter taps from the banded matrices: row 2 of the sliced
  // band matrix holds REC[j] at column (2*2 + j - 3) = j + 1, j = 0..7.
  __shared__ float s_coef[16];
  const int tid = threadIdx.x;
  if (tid < 8) {
    s_coef[tid]     = mlo[(size_t)2 * L + 1 + tid];
    s_coef[8 + tid] = mhi[(size_t)2 * L + 1 + tid];
  }
  __syncthreads();

  const int wave = tid >> 5;
  const int lane = tid & 31;
  const int hf   = lane >> 4;   // which 16-lane half of the wave
  const int lid  = lane & 15;

  const int k0 = (blockIdx.x * 8 + wave) * 16;  // output-column tile base
  const int r0 = blockIdx.y * 16;               // (n*C + c) row tile base
  if (k0 >= Kout || r0 >= R) return;            // wave-uniform: EXEC stays full

  // ---- Build B fragments (K x N = 4 x 16, f32). Wave32 layout: lane
  // (hf, lid), register v holds element (K = v + 2*hf, N = lid).
  // B[l_local, n] = M[k0+n, lbase0 + 4c + l_local] = REC[j],
  //   j = 4c + (v + 2*hf) - 2*lid - 1, zero outside [0, 8).
  v2f blo[10], bhi[10];
#pragma unroll
  for (int c = 0; c < 10; ++c) {
#pragma unroll
    for (int v = 0; v < 2; ++v) {
      int  j  = 4 * c + (v + 2 * hf) - 2 * lid - 1;
      bool ok = (unsigned)j < 8u;
      int  jj = ok ? j : 0;
      blo[c][v] = ok ? s_coef[jj]     : 0.0f;
      bhi[c][v] = ok ? s_coef[8 + jj] : 0.0f;
    }
  }

  // ---- A fragments: X[r0+lid, lbase0 + 4c + (v + 2*hf)], lbase0 = 2*k0 - 4.
  const float* xrow   = x + (size_t)(r0 + lid) * L;
  const int    lbase  = 2 * k0 - 4 + 2 * hf;  // this lane's pair base
  const int    lbase0 = 2 * k0 - 4;           // wave-uniform tile l-range base

  v2f a[10];
  if (lbase0 >= 0 && lbase0 + 39 < L) {
    // Interior tile (all but first/last column tile): unpredicated batch of
    // 10 b64 loads from one address with immediate offsets -> full MLP.
#pragma unroll
    for (int c = 0; c < 10; ++c) a[c] = *(const v2f*)(xrow + lbase + 4 * c);
  } else {
    // Edge tile: zero out-of-range pairs (reproduces band-matrix truncation).
#pragma unroll
    for (int c = 0; c < 10; ++c) {
      int lp = lbase + 4 * c;        // even; a pair never straddles an edge
      v2f t = {0.0f, 0.0f};
      if (lp >= 0 && lp + 1 < L) t = *(const v2f*)(xrow + lp);
      a[c] = t;
    }
  }

  // ---- D = sum_c A_c(16x4) * B_c(4x16) + C, f32 accumulate. Alternating
  // lo/hi accumulator chains: no D->A/B hazard between consecutive WMMAs.
  v8f alo = {};
  v8f ahi = {};
#pragma unroll
  for (int c = 0; c < 10; ++c) {
    alo = __builtin_amdgcn_wmma_f32_16x16x4_f32(false, a[c], false, blo[c],
                                                (short)0, alo, false, false);
    ahi = __builtin_amdgcn_wmma_f32_16x16x4_f32(false, a[c], false, bhi[c],
                                                (short)0, ahi, false, false);
  }

  // ---- Store D. Wave32 f32 C/D layout: VGPR v, lanes 0-15 -> M=v,
  // lanes 16-31 -> M=v+8; N = lid. lo first, hi second (tuple return order).
  float* olo = out;
  float* ohi = out + (size_t)R * Kout;
  const int kk = k0 + lid;
#pragma unroll
  for (int v = 0; v < 8; ++v) {
    size_t idx = (size_t)(r0 + v + 8 * hf) * Kout + kk;
    olo[idx] = alo[v];
    ohi[idx] = ahi[v];
  }
}

extern "C" void kernel_launch(void* const* d_in, const int* in_sizes, int n_in,
                              void* d_out, int out_size, void* d_ws, size_t ws_size,
                              hipStream_t stream) {
  const float* x   = (const float*)d_in[0];  // (N, C, L) f32
  const float* mlo = (const float*)d_in[1];  // (L/2, L) f32 banded
  const float* mhi = (const float*)d_in[2];  // (L/2, L) f32 banded
  float*       out = (float*)d_out;          // lo (R, L/2) then hi (R, L/2)

  // in_sizes[1] = (L/2)*L  ->  L = sqrt(2*in_sizes[1]);  R = in_sizes[0]/L
  const long long me = (long long)in_sizes[1];
  const int L    = (int)llround(sqrt(2.0 * (double)me));   // 8192
  const int R    = in_sizes[0] / L;                        // 1024
  const int Kout = L / 2;                                  // 4096

  const int colTiles = (Kout + 15) / 16;       // 256
  dim3 block(256);                             // 8 waves, 1 tile each
  dim3 grid((colTiles + 7) / 8, (R + 15) / 16);
  dwt1d_wmma<<<grid, block, 0, stream>>>(x, mlo, mhi, out, R, L, Kout);
}